// Huckel_45638322487560
// MI455X (gfx1250) — compile-verified
//
#include <hip/hip_runtime.h>
#include <hip/hip_bf16.h>

typedef __attribute__((ext_vector_type(2))) float v2f;
typedef __attribute__((ext_vector_type(8))) float v8f;

#define N 128
#define NN (N * N)          // 16384
#define NTHREADS 256
#define NWAVES 8
#define NITER 40

// ---------------------------------------------------------------------------
// 128x128x128 f32 GEMM, all operands in LDS, using V_WMMA_F32_16X16X4_F32.
// Register-blocked: each wave owns one 16-wide column strip (tn = wave) and
// all 8 row-tiles of it simultaneously (8 x v8f accumulators).  Per K-step of
// 4: one B-fragment load (shared by all 8 row tiles) + 8 A-fragment loads +
// 8 WMMAs -> 1.25 DS ops per WMMA instead of 3.
// A-frag layout (ISA 7.12.2, 32-bit A 16x4): lanes 0-15 -> M=lane, K=0/1;
// lanes 16-31 -> M=lane-16, K=2/3.  B/C: rows striped across lanes per VGPR.
// ---------------------------------------------------------------------------
__device__ __forceinline__ void gemm128(const float* __restrict__ A,
                                        const float* __restrict__ B,
                                        float* __restrict__ C,
                                        int wave, int lane) {
    const int half = lane >> 4;     // 0: lanes 0-15, 1: lanes 16-31
    const int l15  = lane & 15;
    const int tn   = wave;          // this wave's column strip

    v8f acc[8];
    #pragma unroll
    for (int m = 0; m < 8; ++m)
        acc[m] = (v8f){0.f, 0.f, 0.f, 0.f, 0.f, 0.f, 0.f, 0.f};

    const float* Bcol  = B + (half * 2) * N + tn * 16 + l15;  // rows K0/K2 of strip
    const float* Abase = A + l15 * N + half * 2;              // K offsets 0/1 or 2/3

    #pragma unroll 2
    for (int kk = 0; kk < N; kk += 4) {
        v2f bf;                                   // one B fragment, reused 8x
        bf.x = Bcol[kk * N];                      // B[kk+kb  ][n]
        bf.y = Bcol[kk * N + N];                  // B[kk+kb+1][n]
        #pragma unroll
        for (int m = 0; m < 8; ++m) {
            v2f af = *(const v2f*)(Abase + (m * 16) * N + kk);
            acc[m] = __builtin_amdgcn_wmma_f32_16x16x4_f32(
                false, af, false, bf, (short)0, acc[m], false, false);
        }
    }

    // C layout: VGPR r -> row (m*16 + r + half*8), col (tn*16 + l15)
    #pragma unroll
    for (int m = 0; m < 8; ++m) {
        float* Crow = C + (m * 16 + half * 8) * N + tn * 16 + l15;
        #pragma unroll
        for (int r = 0; r < 8; ++r) Crow[r * N] = acc[m][r];
    }
}

// ---------------------------------------------------------------------------
// One workgroup per molecule: scatter-build H in LDS, symmetrize from tril,
// Palser-Manolopoulos canonical purification (pure GEMMs), energy = 2 tr(XH).
// ---------------------------------------------------------------------------
__global__ __launch_bounds__(NTHREADS)
void huckel_purify_kernel(const float* __restrict__ weights,
                          const int*   __restrict__ ham_idx,
                          const int*   __restrict__ par_idx,
                          const int*   __restrict__ n_elec,
                          float* __restrict__ out,
                          int E) {
    extern __shared__ float smem[];
    float* H   = smem;              // 16384 f32
    float* X   = H  + NN;           // 16384
    float* X2  = X  + NN;           // 16384
    float* X3  = X2 + NN;           // 16384
    float* red = X3 + NN;           // 256 reduction scratch
    float* sc  = red + 256;         // 16 broadcast scalars

    const int b    = blockIdx.x;
    const int tid  = threadIdx.x;
    const int wave = tid >> 5;
    const int lane = tid & 31;

    // ---- phase 0: zero H --------------------------------------------------
    for (int i = tid; i < NN; i += NTHREADS) H[i] = 0.0f;
    __syncthreads();

    // ---- phase 1: scatter-add entries into H (LDS float atomics) ----------
    {
        const long base = (long)b * E;
        for (int e = tid; e < E; e += NTHREADS) {
            const int r = ham_idx[(base + e) * 2 + 0];
            const int c = ham_idx[(base + e) * 2 + 1];
            const float v = weights[par_idx[base + e]];
            atomicAdd(&H[r * N + c], v);   // ds_add_f32
        }
    }
    __syncthreads();

    // ---- phase 2: symmetrize from lower triangle (tril + tril^T + diag) ---
    for (int idx = tid; idx < NN; idx += NTHREADS) {
        const int i = idx >> 7, j = idx & 127;
        if (i < j) H[i * N + j] = H[j * N + i];   // writes only upper, reads only lower
    }
    __syncthreads();

    // ---- phase 3: Gershgorin bounds + trace -> purification seed ----------
    if (tid < N) {
        const float* row = H + tid * N;
        const float d = row[tid];
        float rad = 0.0f;
        for (int j = 0; j < N; ++j) rad += fabsf(row[j]);
        rad -= fabsf(d);
        red[tid]       = d - rad;   // row lower bound
        red[128 + tid] = d + rad;   // row upper bound
    }
    __syncthreads();
    if (tid == 0) {
        float emin = red[0], emax = red[128], trH = 0.0f;
        for (int i = 0; i < N; ++i) {
            emin = fminf(emin, red[i]);
            emax = fmaxf(emax, red[128 + i]);
            trH += H[i * N + i];
        }
        const int   k     = n_elec[b] >> 1;              // doubly-occupied orbitals
        const float theta = (float)k / (float)N;         // filling fraction
        const float mu    = trH / (float)N;
        const float lam   = fminf(theta        / fmaxf(emax - mu, 1e-6f),
                                  (1.f - theta) / fmaxf(mu - emin, 1e-6f));
        sc[1] = lam; sc[2] = mu; sc[3] = theta;
    }
    __syncthreads();

    // ---- phase 4: X0 = lam*(mu*I - H) + theta*I ---------------------------
    {
        const float lam = sc[1], mu = sc[2], theta = sc[3];
        for (int idx = tid; idx < NN; idx += NTHREADS) {
            const int i = idx >> 7, j = idx & 127;
            const float diag = (i == j) ? 1.0f : 0.0f;
            X[idx] = lam * (mu * diag - H[idx]) + theta * diag;
        }
    }
    __syncthreads();

    // ---- phase 5: canonical purification (all WMMA GEMMs) -----------------
    #pragma unroll 1
    for (int it = 0; it < NITER; ++it) {
        gemm128(X,  X, X2, wave, lane);   // X2 = X*X
        __syncthreads();
        gemm128(X2, X, X3, wave, lane);   // X3 = X*X*X
        __syncthreads();

        // trace-conserving mixing coefficient c = tr(X2 - X3)/tr(X - X2)
        if (tid < N) {
            const float tx  = X [tid * N + tid];
            const float tx2 = X2[tid * N + tid];
            const float tx3 = X3[tid * N + tid];
            red[tid]       = tx  - tx2;
            red[128 + tid] = tx2 - tx3;
        }
        __syncthreads();
        if (tid == 0) {
            float den = 0.f, num = 0.f;
            for (int i = 0; i < N; ++i) { den += red[i]; num += red[128 + i]; }
            float c = (fabsf(den) > 1e-9f) ? (num / den) : 0.5f;  // converged -> McWeeny
            c = fminf(fmaxf(c, 1e-3f), 1.f - 1e-3f);
            sc[0] = c;
        }
        __syncthreads();

        const float c = sc[0];
        if (c >= 0.5f) {
            const float inv = 1.0f / (1.0f - c);
            for (int idx = tid; idx < NN; idx += NTHREADS)
                X[idx] = ((1.f - 2.f * c) * X[idx] + (1.f + c) * X2[idx] - X3[idx]) * inv;
        } else {
            const float inv = 1.0f / c;
            for (int idx = tid; idx < NN; idx += NTHREADS)
                X[idx] = ((1.f + c) * X2[idx] - X3[idx]) * inv;
        }
        __syncthreads();
    }

    // ---- phase 6: energy = 2 * tr(X * H) ----------------------------------
    {
        float part = 0.0f;
        for (int idx = tid; idx < NN; idx += NTHREADS) part += X[idx] * H[idx];
        red[tid] = part;
    }
    __syncthreads();
    if (tid == 0) {
        float s = 0.0f;
        for (int i = 0; i < NTHREADS; ++i) s += red[i];
        out[b] = 2.0f * s;
    }
}

// ---------------------------------------------------------------------------
extern "C" void kernel_launch(void* const* d_in, const int* in_sizes, int n_in,
                              void* d_out, int out_size, void* d_ws, size_t ws_size,
                              hipStream_t stream) {
    const float* weights = (const float*)d_in[0];   // (6,7,5,5) f32
    const int*   ham_idx = (const int*)d_in[1];     // (B,E,2) i32
    const int*   par_idx = (const int*)d_in[2];     // (B,E)   i32
    const int*   n_elec  = (const int*)d_in[3];     // (B,)    i32
    float*       out     = (float*)d_out;           // (B,)    f32

    const int B = in_sizes[3];
    const int E = in_sizes[2] / B;

    const size_t shmem = (size_t)(4 * NN + 256 + 16) * sizeof(float); // 263232 B < 320 KB
    huckel_purify_kernel<<<dim3(B), dim3(NTHREADS), shmem, stream>>>(
        weights, ham_idx, par_idx, n_elec, out, E);
}